// RelativePositionEncoding_61856118997302
// MI455X (gfx1250) — compile-verified
//
#include <hip/hip_runtime.h>
#include <stdint.h>

// Problem constants (fixed by the reference).
#define SEQ_LEN   262144
#define E_DIMS    256
#define ATTRS     8
#define PATTERN_BYTES (ATTRS * E_DIMS * 4)      // 8192 B repeating tile

// Launch geometry: 1024 blocks x 256 threads (8 wave32 per block).
// Each block writes a contiguous 256 KiB chunk = 32 pattern periods,
// so every chunk base is pattern-aligned (262144 % 8192 == 0).
#define NUM_BLOCKS        1024
#define THREADS_PER_BLOCK 256
#define BYTES_PER_BLOCK   (SEQ_LEN * E_DIMS * 4 / NUM_BLOCKS)   // 262144 B
#define BYTES_PER_WAVE    (BYTES_PER_BLOCK / 8)                 // 32768 B
#define ITERS_PER_WAVE    (BYTES_PER_WAVE / 512)                // 64 async stores

__global__ __launch_bounds__(THREADS_PER_BLOCK)
void rpe_broadcast_kernel(const float* __restrict__ E, float* __restrict__ out) {
    // 8 KB repeating tile staged in LDS: pattern[row%8][col] = E[row%8][col].
    __shared__ float pattern[ATTRS * E_DIMS];   // 2048 floats = 8192 bytes

    const int tid = threadIdx.x;

    // Stage E (8x256 fp32 = 512 float4) into LDS: 2 float4 loads per thread.
    const float4* __restrict__ E4 = reinterpret_cast<const float4*>(E);
    float4* P4 = reinterpret_cast<float4*>(pattern);
    P4[tid]       = E4[tid];
    P4[tid + 256] = E4[tid + 256];
    __syncthreads();

    const unsigned lane = tid & 31u;
    const unsigned wave = (unsigned)tid >> 5;

    // SGPR base address of this block's 256 KiB output chunk (GVS mode saddr).
    const char* base_ptr = reinterpret_cast<const char*>(out)
                         + (unsigned long long)blockIdx.x * (unsigned long long)BYTES_PER_BLOCK;

    // LDS byte offset of the pattern tile within this block's LDS allocation
    // (flat LDS addresses carry the allocation-relative offset in the low 32 bits).
    const unsigned lds_base = (unsigned)(uintptr_t)(void*)pattern;

    // Per-lane running global byte offset within the chunk.
    unsigned voff = wave * (unsigned)BYTES_PER_WAVE + lane * 16u;
    const unsigned lane_lds = lds_base + lane * 16u;

    // Stream 64 x 512 B async LDS->global stores per wave, non-temporal:
    // the 256 MiB output is a single-pass stream larger than the 192 MB L2,
    // so bypass temporal residency and head straight for HBM.
    // Each issue: 32 lanes x b128 = 512 B; source LDS offset = (global off) mod 8192.
    #pragma unroll
    for (int i = 0; i < ITERS_PER_WAVE; ++i) {
        const unsigned lds_off = lane_lds + (unsigned)((i & 15) << 9);  // (i%16)*512
        asm volatile(
            "global_store_async_from_lds_b128 %0, %1, %2 th:TH_STORE_NT"
            :
            : "v"(voff), "v"(lds_off), "s"(base_ptr)
            : "memory");
        voff += 512u;
    }

    // Drain ASYNCcnt before the wave retires (also keeps LDS live until done).
    asm volatile("s_wait_asynccnt 0x0" ::: "memory");
}

extern "C" void kernel_launch(void* const* d_in, const int* in_sizes, int n_in,
                              void* d_out, int out_size, void* d_ws, size_t ws_size,
                              hipStream_t stream) {
    (void)in_sizes; (void)n_in; (void)out_size; (void)d_ws; (void)ws_size;
    // d_in[0] = x (int32, values unused by the module; only its length matters)
    // d_in[1] = E_relative_position (fp32, 8 x 256)
    const float* E = reinterpret_cast<const float*>(d_in[1]);
    float* out = reinterpret_cast<float*>(d_out);

    rpe_broadcast_kernel<<<NUM_BLOCKS, THREADS_PER_BLOCK, 0, stream>>>(E, out);
}